// ConsciousnessMetrics_57715770524288
// MI455X (gfx1250) — compile-verified
//
#include <hip/hip_runtime.h>
#include <math.h>

// ---------------------------------------------------------------------------
// Problem constants
// ---------------------------------------------------------------------------
#define NROWS 8192
#define NCOLS 4096
#define ROWBLOCKS 64               // kernel B splits rows into 64 blocks of 128
#define ROWS_PER_RB 128
#define NPAIRS_TOTAL 33546240.0f   // 8192 * 4095

typedef __attribute__((ext_vector_type(2))) float v2f;
typedef __attribute__((ext_vector_type(4))) float v4f;
typedef __attribute__((ext_vector_type(8))) float v8f;

// ---------------------------------------------------------------------------
// Exact f32 wave(32) sum using two chained V_WMMA_F32_16X16X4_F32, no shuffle.
//
// A-matrix 16x4 f32 layout: lane m holds A[m][0..1] in its 2 VGPRs, lane m+16
// holds A[m][2..3]. With A = {v, 0} and B = ones(4x16):
//   D[m][n] = v_m + v_{m+16}  (for every n)
// D VGPR j = D[j][lane] (lanes 0-15) / D[j+8][lane-16] (lanes 16-31), so
// r = Sum_j d[j] gives each 16-lane half its half-total (r_lo / r_hi).
// A second WMMA with A = {r, 0} yields D'[m][n] = r_m + r_{m+16} = r_lo + r_hi
// = the full 32-lane sum, broadcast to ALL lanes in d'[0]. Pure f32 RNE adds.
// ---------------------------------------------------------------------------
__device__ __forceinline__ float wave_sum32(float v) {
  v2f b; b[0] = 1.0f; b[1] = 1.0f;
  v2f a; a[0] = v;    a[1] = 0.0f;
  v8f c = {};
  c = __builtin_amdgcn_wmma_f32_16x16x4_f32(
      false, a, false, b, (short)0, c, false, false);
  const float r = ((c[0] + c[1]) + (c[2] + c[3])) +
                  ((c[4] + c[5]) + (c[6] + c[7]));
  v2f a2; a2[0] = r; a2[1] = 0.0f;
  v8f c2 = {};
  c2 = __builtin_amdgcn_wmma_f32_16x16x4_f32(
      false, a2, false, b, (short)0, c2, false, false);
  return c2[0];
}

// Batched block (256 threads = 8 waves) sums: N quantities, one barrier pair.
// Results broadcast to all threads (in-place). Uniform control flow required.
template <int N>
__device__ __forceinline__ void block_sum_n(float (&v)[N], float* red /*N*8*/) {
  const int lane = threadIdx.x & 31;
  const int wid  = threadIdx.x >> 5;
  #pragma unroll
  for (int q = 0; q < N; ++q) {
    const float s = wave_sum32(v[q]);
    if (lane == 0) red[q * 8 + wid] = s;
  }
  __syncthreads();
  #pragma unroll
  for (int q = 0; q < N; ++q) {
    const float* r = red + q * 8;
    v[q] = ((r[0] + r[1]) + (r[2] + r[3])) + ((r[4] + r[5]) + (r[6] + r[7]));
  }
  __syncthreads();
}

// ---------------------------------------------------------------------------
// Kernel 0: zero the 64-bin global pair histogram (ws is poisoned by harness)
// ---------------------------------------------------------------------------
__global__ void init_hist(unsigned int* __restrict__ ghist) {
  if (threadIdx.x < 64) ghist[threadIdx.x] = 0u;
}

// ---------------------------------------------------------------------------
// Kernel A: one block per row. Streams psi row (entropy, sum, sumsq, pair
// histogram) and field row (sum, sumsq) exactly once. 256 thr x 16 elems.
// psi loads are NON-TEMPORAL (read-once) so the field tensor (read again by
// col_pass, 128 MB < 192 MB L2) stays L2-resident.
// ---------------------------------------------------------------------------
__global__ __launch_bounds__(256) void row_pass(
    const float* __restrict__ psi, const float* __restrict__ fld,
    unsigned int* __restrict__ ghist,
    float* __restrict__ entArr, float* __restrict__ stdArr,
    float* __restrict__ nrmArr, float* __restrict__ fsArr,
    float* __restrict__ fssArr)
{
  __shared__ unsigned int shist[64];
  __shared__ float red[40];
  const int t   = threadIdx.x;
  const int row = blockIdx.x;
  if (t < 64) shist[t] = 0u;
  __syncthreads();

  const float* prow = psi + (size_t)row * NCOLS;
  const float* frow = fld + (size_t)row * NCOLS;
  const int base = t << 4;   // 16 elements per thread

  float s1 = 0.f, s2 = 0.f, se = 0.f;
  int qprev = 0;
  #pragma unroll
  for (int k = 0; k < 4; ++k) {
    const v4f p4 = __builtin_nontemporal_load((const v4f*)(prow + base + 4 * k));
    #pragma unroll
    for (int j = 0; j < 4; ++j) {
      const float p = p4[j];
      s1 += p;
      s2 += p * p;
      se += p * __logf(p + 1e-10f);           // entropy term (saturates d_eeg cap)
      int q = (int)floorf(p * 8.0f);
      q = q < 0 ? 0 : (q > 7 ? 7 : q);
      if (k | j) atomicAdd(&shist[(qprev << 3) + q], 1u);  // 15 intra-chunk pairs
      qprev = q;
    }
  }
  // boundary pair (elem base+15, elem base+16) -> next thread's first element
  if (t < 255) {
    const float pn = __builtin_nontemporal_load(prow + base + 16);
    int qn = (int)floorf(pn * 8.0f);
    qn = qn < 0 ? 0 : (qn > 7 ? 7 : qn);
    atomicAdd(&shist[(qprev << 3) + qn], 1u);
  }
  // => 256*15 + 255 = 4095 pairs per row, matching B*(E-1)/B

  float fs = 0.f, fss = 0.f;
  #pragma unroll
  for (int k = 0; k < 4; ++k) {
    const float4 f4 = *(const float4*)(frow + base + 4 * k);  // RT: keep in L2
    fs  += (f4.x + f4.y) + (f4.z + f4.w);
    fss += (f4.x * f4.x + f4.y * f4.y) + (f4.z * f4.z + f4.w * f4.w);
  }

  // uniform control flow, all lanes active -> WMMA reductions legal
  float sums[5] = {s1, s2, se, fs, fss};
  block_sum_n<5>(sums, red);

  if (t < 64) atomicAdd(&ghist[t], shist[t]);   // integer atomics: deterministic

  if (t == 0) {
    const float S1 = sums[0], S2 = sums[1], SE = sums[2];
    const float FS = sums[3], FSS = sums[4];
    entArr[row] = -SE;                                            // row entropy
    const float var = (S2 - S1 * S1 * (1.0f / (float)NCOLS)) *
                      (1.0f / (float)(NCOLS - 1));                // ddof=1
    stdArr[row] = sqrtf(fmaxf(var, 0.0f));
    nrmArr[row] = sqrtf(FSS);                                     // row L2 norm
    fsArr[row]  = FS;
    fssArr[row] = FSS;
  }
}

// ---------------------------------------------------------------------------
// Kernel B: column-oriented pass over field (mostly L2 hits after row_pass).
// Thread t of block (bx, rb) owns column c = bx*256+t over 128 rows; fully
// coalesced main load, neighbor load hits the same/next cacheline.
// ---------------------------------------------------------------------------
__global__ __launch_bounds__(256) void col_pass(
    const float* __restrict__ fld,
    float* __restrict__ pSx, float* __restrict__ pSxx, float* __restrict__ pSxy)
{
  const int c  = blockIdx.x * 256 + threadIdx.x;     // 0..4095
  const int rb = blockIdx.y;                         // 0..63
  const int cn = c < (NCOLS - 1) ? c + 1 : (NCOLS - 1);
  const float* p = fld + (size_t)rb * ROWS_PER_RB * NCOLS;
  float sx = 0.f, sxx = 0.f, sxy = 0.f;
  #pragma unroll 4
  for (int i = 0; i < ROWS_PER_RB; ++i) {
    const float v  = p[c];
    const float vn = p[cn];
    sx  += v;
    sxx += v * v;
    sxy += v * vn;
    p += NCOLS;
  }
  const size_t o = (size_t)rb * NCOLS + c;
  pSx[o] = sx; pSxx[o] = sxx; pSxy[o] = sxy;         // sxy[4095] ignored later
}

// ---------------------------------------------------------------------------
// Kernel C: single-block finalize. Deterministic tree reductions everywhere.
// ---------------------------------------------------------------------------
__global__ __launch_bounds__(256) void finalize(
    const float* __restrict__ entArr, const float* __restrict__ stdArr,
    const float* __restrict__ nrmArr, const float* __restrict__ fsArr,
    const float* __restrict__ fssArr,
    const float* __restrict__ pSx, const float* __restrict__ pSxx,
    const float* __restrict__ pSxy,
    float* __restrict__ colSx, float* __restrict__ colSxx,
    float* __restrict__ colSxy,
    const unsigned int* __restrict__ ghist,
    const float* __restrict__ w, float* __restrict__ out)
{
  __shared__ float red[40];
  const int t = threadIdx.x;

  // ---- reduce per-row arrays -------------------------------------------
  float a_ent = 0.f, a_std = 0.f, a_nrm = 0.f, a_fs = 0.f, a_fss = 0.f;
  for (int i = t; i < NROWS; i += 256) {
    a_ent += entArr[i]; a_std += stdArr[i]; a_nrm += nrmArr[i];
    a_fs  += fsArr[i];  a_fss += fssArr[i];
  }
  float rowSums[5] = {a_ent, a_std, a_nrm, a_fs, a_fss};
  block_sum_n<5>(rowSums, red);
  const float sum_ent = rowSums[0], sum_std = rowSums[1], sum_nrm = rowSums[2];
  const float gsum = rowSums[3], gss = rowSums[4];

  // ---- column totals over the 64 row-block partials --------------------
  for (int c = t; c < NCOLS; c += 256) {
    float sx = 0.f, sxx = 0.f, sxy = 0.f;
    for (int rb = 0; rb < ROWBLOCKS; ++rb) {
      const size_t o = (size_t)rb * NCOLS + c;
      sx += pSx[o]; sxx += pSxx[o]; sxy += pSxy[o];
    }
    colSx[c] = sx; colSxx[c] = sxx; colSxy[c] = sxy;
  }
  __syncthreads();

  // ---- adjacent-column Pearson correlation -----------------------------
  float accC = 0.f, cnt = 0.f;
  for (int c = t; c < NCOLS - 1; c += 256) {
    const float sx0 = colSx[c], sx1 = colSx[c + 1];
    const float invB = 1.0f / (float)NROWS;
    const float var0 = colSxx[c]     - sx0 * sx0 * invB;
    const float var1 = colSxx[c + 1] - sx1 * sx1 * invB;
    const float cov  = colSxy[c]     - sx0 * sx1 * invB;
    const float corr = cov / sqrtf(var0 * var1);
    if (!(corr != corr)) { accC += corr; cnt += 1.0f; }   // skip NaN only
  }
  float cSums[2] = {accC, cnt};
  block_sum_n<2>(cSums, red);
  const float sumCorr = cSums[0], nCorr = cSums[1];

  // ---- histogram conditional entropy -----------------------------------
  float hterm[1] = {0.f};
  if (t < 64) {
    const float p = (float)ghist[t] / NPAIRS_TOTAL;
    if (p > 0.f) hterm[0] = -p * log2f(p);
  }
  block_sum_n<1>(hterm, red);
  const float cond_ent = hterm[0];

  if (t == 0) {
    const float entropy = sum_ent / (float)NROWS;
    const float mstd    = sum_std / (float)NROWS;
    const float d_eeg   = fminf(entropy * mstd * 3.0f, 10.0f);

    const float fmag      = sum_nrm / (float)NROWS;
    const float mean_corr = (nCorr > 0.f) ? (sumCorr / fmaxf(nCorr, 1.0f)) : 0.0f;
    const float h_fmri    = fminf(fmag * fabsf(mean_corr) * 2.0f, 5.0f);

    const double n    = (double)NROWS * (double)NCOLS;
    const double gvar = ((double)gss - (double)gsum * (double)gsum / n) / (n - 1.0);
    const float  fcomp = sqrtf(fmaxf((float)gvar, 0.0f));
    const float  clz   = fminf(cond_ent + 0.3f * fcomp, 3.0f);

    const float fci = w[0] * (d_eeg * 0.1f) + w[1] * (h_fmri * 0.2f) +
                      w[2] * (clz * (1.0f / 3.0f));
    out[0] = fminf(fmaxf(fci, 0.0f), 1.0f);     // / D_MAX (=1) then clip
  }
}

// ---------------------------------------------------------------------------
// Launch
// ---------------------------------------------------------------------------
extern "C" void kernel_launch(void* const* d_in, const int* in_sizes, int n_in,
                              void* d_out, int out_size, void* d_ws, size_t ws_size,
                              hipStream_t stream) {
  (void)in_sizes; (void)n_in; (void)out_size; (void)ws_size;
  const float* psi = (const float*)d_in[0];
  const float* fld = (const float*)d_in[1];
  const float* w   = (const float*)d_in[2];
  float* out = (float*)d_out;

  // workspace layout (floats): hist(64 u32) | 5 row arrays | 3 col-partial
  // planes (64*4096 each) | 3 col-total arrays  => ~3.3 MB total
  float* ws = (float*)d_ws;
  unsigned int* hist = (unsigned int*)ws;
  float* entArr = ws + 64;
  float* stdArr = entArr + NROWS;
  float* nrmArr = stdArr + NROWS;
  float* fsArr  = nrmArr + NROWS;
  float* fssArr = fsArr  + NROWS;
  float* pSx    = fssArr + NROWS;
  float* pSxx   = pSx  + (size_t)ROWBLOCKS * NCOLS;
  float* pSxy   = pSxx + (size_t)ROWBLOCKS * NCOLS;
  float* colSx  = pSxy + (size_t)ROWBLOCKS * NCOLS;
  float* colSxx = colSx  + NCOLS;
  float* colSxy = colSxx + NCOLS;

  hipLaunchKernelGGL(init_hist, dim3(1), dim3(64), 0, stream, hist);
  hipLaunchKernelGGL(row_pass, dim3(NROWS), dim3(256), 0, stream,
                     psi, fld, hist, entArr, stdArr, nrmArr, fsArr, fssArr);
  hipLaunchKernelGGL(col_pass, dim3(NCOLS / 256, ROWBLOCKS), dim3(256), 0, stream,
                     fld, pSx, pSxx, pSxy);
  hipLaunchKernelGGL(finalize, dim3(1), dim3(256), 0, stream,
                     entArr, stdArr, nrmArr, fsArr, fssArr,
                     pSx, pSxx, pSxy, colSx, colSxx, colSxy, hist, w, out);
}